// GCN_6751688589931
// MI455X (gfx1250) — compile-verified
//
#include <hip/hip_runtime.h>

// ---------------------------------------------------------------------------
// GraphConv GNN forward for MI455X (gfx1250).
//   BN -> [edge scatter + WMMA-f32 GEMM + ReLU (not layer 3)] x3 -> mean-pool -> linear
// CDNA5 paths used:
//   * v_wmma_f32_16x16x4_f32 (full-precision matrix core, K accumulated 4-wide)
//   * global_load_async_to_lds_b128 + s_wait_asynccnt (async weight staging)
//   * ds broadcast reads for B fragments
// ---------------------------------------------------------------------------

typedef __attribute__((ext_vector_type(2))) float v2f;
typedef __attribute__((ext_vector_type(8))) float v8f;

#define N_NODES   80000
#define N_EDGES   1280000
#define N_FEAT    4
#define HIDDEN    64
#define N_GRAPHS  64
#define BN_EPS    1e-5f

static_assert(N_NODES % 16 == 0, "row tiles must divide exactly");

// ---------------------------------------------------------------- utilities

__global__ void zero_kernel(float* __restrict__ p, int n) {
    int i = blockIdx.x * blockDim.x + threadIdx.x;
    if (i < n) p[i] = 0.0f;
}

// ------------------------------------------------------------- BatchNorm1d

__global__ void bn_kernel(const float* __restrict__ x,
                          const float* __restrict__ gamma,
                          const float* __restrict__ beta,
                          const float* __restrict__ mean,
                          const float* __restrict__ var,
                          float* __restrict__ h0) {
    int i = blockIdx.x * blockDim.x + threadIdx.x;
    if (i >= N_NODES * N_FEAT) return;
    int c = i & 3;
    float scale = gamma[c] * rsqrtf(var[c] + BN_EPS);
    h0[i] = (x[i] - mean[c]) * scale + beta[c];
}

// -------------------------------------------------------- edge scatter-add

// layer 1: 4 features per node, one thread per edge
__global__ void scatter4_kernel(const float* __restrict__ h,
                                const int* __restrict__ src,
                                const int* __restrict__ dst,
                                float* __restrict__ agg) {
    int e = blockIdx.x * blockDim.x + threadIdx.x;
    if (e >= N_EDGES) return;
    int s = src[e], d = dst[e];
    float4 v = *(const float4*)(h + (long long)s * 4);
    float* a = agg + (long long)d * 4;
    atomicAdd(a + 0, v.x);
    atomicAdd(a + 1, v.y);
    atomicAdd(a + 2, v.z);
    atomicAdd(a + 3, v.w);
}

// layers 2/3: 64 features, 4 threads per edge (16 features each)
__global__ void scatter64_kernel(const float* __restrict__ h,
                                 const int* __restrict__ src,
                                 const int* __restrict__ dst,
                                 float* __restrict__ agg) {
    long long t = (long long)blockIdx.x * blockDim.x + threadIdx.x;
    if (t >= (long long)N_EDGES * 4) return;
    int e   = (int)(t >> 2);
    int c16 = (int)(t & 3) * 16;
    int s = src[e], d = dst[e];
    const float* hp = h   + (long long)s * HIDDEN + c16;
    float*       ap = agg + (long long)d * HIDDEN + c16;
#pragma unroll
    for (int q = 0; q < 4; ++q) {
        float4 v = *(const float4*)(hp + q * 4);
        atomicAdd(ap + q * 4 + 0, v.x);
        atomicAdd(ap + q * 4 + 1, v.y);
        atomicAdd(ap + q * 4 + 2, v.z);
        atomicAdd(ap + q * 4 + 3, v.w);
    }
}

// --------------------------------------------------- WMMA GraphConv (K = 4)
// out = relu(agg @ Wrel + brel + h @ Wroot); C_in = 4 -> one 16x16x4 WMMA
// per matrix covers all of K.  One 16x16 output tile per wave32.

__global__ void conv_gemm4_kernel(const float* __restrict__ agg,
                                  const float* __restrict__ h,
                                  const float* __restrict__ Wrel,
                                  const float* __restrict__ brel,
                                  const float* __restrict__ Wroot,
                                  float* __restrict__ out) {
    int wave = threadIdx.x >> 5;
    int lane = threadIdx.x & 31;
    int tile = blockIdx.x * 8 + wave;          // 8 waves / block, exact grid
    int rowTile = tile >> 2;                   // 5000 row tiles
    int colTile = tile & 3;                    // 4 col tiles (HIDDEN/16)
    int row0 = rowTile * 16, col0 = colTile * 16;

    int m    = lane & 15;                      // A-matrix row within tile
    int kb   = (lane >> 4) * 2;                // K base: lanes<16 -> 0, else 2
    int nCol = lane & 15;                      // B/C column within tile

    v2f aR, aH, bR, bH;
    const float* ar = agg + (long long)(row0 + m) * N_FEAT + kb;
    aR[0] = ar[0]; aR[1] = ar[1];
    const float* ah = h + (long long)(row0 + m) * N_FEAT + kb;
    aH[0] = ah[0]; aH[1] = ah[1];
    bR[0] = Wrel [(kb + 0) * HIDDEN + col0 + nCol];
    bR[1] = Wrel [(kb + 1) * HIDDEN + col0 + nCol];
    bH[0] = Wroot[(kb + 0) * HIDDEN + col0 + nCol];
    bH[1] = Wroot[(kb + 1) * HIDDEN + col0 + nCol];

    v8f c = {};
    c = __builtin_amdgcn_wmma_f32_16x16x4_f32(false, aR, false, bR,
                                              (short)0, c, false, false);
    c = __builtin_amdgcn_wmma_f32_16x16x4_f32(false, aH, false, bH,
                                              (short)0, c, false, false);

    float bias = brel[col0 + nCol];
    int rbase = row0 + (lane >> 4) * 8;        // C layout: VGPR r -> row r/r+8
#pragma unroll
    for (int r = 0; r < 8; ++r) {
        float v = c[r] + bias;
        v = v > 0.0f ? v : 0.0f;               // ReLU
        out[(long long)(rbase + r) * HIDDEN + col0 + nCol] = v;
    }
}

// -------------------------------------------------- WMMA GraphConv (K = 64)
// Both 64x64 weight matrices are staged into LDS once per block with the
// gfx1250 async global->LDS DMA path, then the K-loop reads B fragments as
// conflict-free LDS broadcasts. 32 x v_wmma_f32_16x16x4_f32 per wave.

__global__ void conv_gemm64_kernel(const float* __restrict__ agg,
                                   const float* __restrict__ h,
                                   const float* __restrict__ Wrel,
                                   const float* __restrict__ brel,
                                   const float* __restrict__ Wroot,
                                   float* __restrict__ out,
                                   int do_relu) {
    __shared__ float sW[2 * HIDDEN * HIDDEN];  // [Wrel | Wroot], 32 KB

    // --- async stage: 8192 floats = 2048 x b128, 8 per thread -------------
    {
        unsigned ldsBase = (unsigned)(unsigned long long)(void*)sW;
        int t = threadIdx.x;                   // 256 threads, all lanes active
#pragma unroll
        for (int i = 0; i < 8; ++i) {
            int idx = t + i * 256;             // float4 index in [0, 2048)
            const float* g = (idx < 1024) ? (Wrel + idx * 4)
                                          : (Wroot + (idx - 1024) * 4);
            unsigned lds = ldsBase + idx * 16;
            asm volatile("global_load_async_to_lds_b128 %0, %1, off"
                         :: "v"(lds), "v"((unsigned long long)g)
                         : "memory");
        }
        asm volatile("s_wait_asynccnt 0x0" ::: "memory");
    }
    __syncthreads();

    const float* sWrel  = sW;
    const float* sWroot = sW + HIDDEN * HIDDEN;

    int wave = threadIdx.x >> 5;
    int lane = threadIdx.x & 31;
    int tile = blockIdx.x * 8 + wave;          // exact: 2500 blocks x 8 waves
    int rowTile = tile >> 2;
    int colTile = tile & 3;
    int row0 = rowTile * 16, col0 = colTile * 16;

    int m    = lane & 15;
    int kOff = (lane >> 4) * 2;                // 0 or 2 within each K=4 chunk
    int nCol = lane & 15;

    const float* aggRow = agg + (long long)(row0 + m) * HIDDEN;
    const float* hRow   = h   + (long long)(row0 + m) * HIDDEN;

    v8f c = {};
#pragma unroll
    for (int k0 = 0; k0 < HIDDEN; k0 += 4) {
        int kb = k0 + kOff;
        v2f a, b;
        // rel path: agg @ Wrel  (B from LDS broadcast)
        a[0] = aggRow[kb]; a[1] = aggRow[kb + 1];
        b[0] = sWrel[(kb + 0) * HIDDEN + col0 + nCol];
        b[1] = sWrel[(kb + 1) * HIDDEN + col0 + nCol];
        c = __builtin_amdgcn_wmma_f32_16x16x4_f32(false, a, false, b,
                                                  (short)0, c, false, false);
        // root path: h @ Wroot
        a[0] = hRow[kb]; a[1] = hRow[kb + 1];
        b[0] = sWroot[(kb + 0) * HIDDEN + col0 + nCol];
        b[1] = sWroot[(kb + 1) * HIDDEN + col0 + nCol];
        c = __builtin_amdgcn_wmma_f32_16x16x4_f32(false, a, false, b,
                                                  (short)0, c, false, false);
    }

    float bias = brel[col0 + nCol];
    int rbase = row0 + (lane >> 4) * 8;
#pragma unroll
    for (int r = 0; r < 8; ++r) {
        float v = c[r] + bias;
        if (do_relu) v = v > 0.0f ? v : 0.0f;
        out[(long long)(rbase + r) * HIDDEN + col0 + nCol] = v;
    }
}

// -------------------------------------------------------- global mean pool

__global__ void pool_kernel(const float* __restrict__ h3,
                            const int* __restrict__ batch,
                            float* __restrict__ sums,
                            float* __restrict__ cnts) {
    int t = blockIdx.x * blockDim.x + threadIdx.x;
    if (t >= N_NODES * HIDDEN) return;
    int node = t >> 6;
    int c    = t & 63;
    int g = batch[node];
    atomicAdd(sums + g * HIDDEN + c, h3[t]);
    if (c == 0) atomicAdd(cnts + g, 1.0f);
}

__global__ void final_kernel(const float* __restrict__ sums,
                             const float* __restrict__ cnts,
                             const float* __restrict__ W,   // [64,2]
                             const float* __restrict__ b,   // [2]
                             float* __restrict__ out) {     // [64,2]
    int t = blockIdx.x * blockDim.x + threadIdx.x;
    if (t >= N_GRAPHS * 2) return;
    int g = t >> 1, cls = t & 1;
    float cnt = cnts[g];
    cnt = cnt > 1.0f ? cnt : 1.0f;
    float acc = 0.0f;
#pragma unroll
    for (int k = 0; k < HIDDEN; ++k)
        acc += (sums[g * HIDDEN + k] / cnt) * W[k * 2 + cls];
    out[t] = acc + b[cls];
}

// ------------------------------------------------------------------ launch

extern "C" void kernel_launch(void* const* d_in, const int* in_sizes, int n_in,
                              void* d_out, int out_size, void* d_ws, size_t ws_size,
                              hipStream_t stream) {
    const float* x       = (const float*)d_in[0];
    const int*   ei      = (const int*)  d_in[1];     // [2, N_EDGES]
    const int*   batch   = (const int*)  d_in[2];
    const float* bn_g    = (const float*)d_in[3];
    const float* bn_b    = (const float*)d_in[4];
    const float* bn_m    = (const float*)d_in[5];
    const float* bn_v    = (const float*)d_in[6];
    const float* W1_rel  = (const float*)d_in[7];
    const float* b1_rel  = (const float*)d_in[8];
    const float* W1_root = (const float*)d_in[9];
    const float* W2_rel  = (const float*)d_in[10];
    const float* b2_rel  = (const float*)d_in[11];
    const float* W2_root = (const float*)d_in[12];
    const float* W3_rel  = (const float*)d_in[13];
    const float* b3_rel  = (const float*)d_in[14];
    const float* W3_root = (const float*)d_in[15];
    const float* lin_W   = (const float*)d_in[16];
    const float* lin_b   = (const float*)d_in[17];
    float* out = (float*)d_out;

    const int* src = ei;
    const int* dst = ei + N_EDGES;

    // workspace: three 80000x64 ping-pong buffers + pool scratch (~62 MB)
    const long long NODEBUF = (long long)N_NODES * HIDDEN;
    float* R0   = (float*)d_ws;            // h (current)
    float* R1   = R0 + NODEBUF;            // agg
    float* R2   = R1 + NODEBUF;            // h (next)
    float* sums = R2 + NODEBUF;            // [64,64]
    float* cnts = sums + N_GRAPHS * HIDDEN;

    const int B = 256;
    const int gemmBlocks = (N_NODES / 16) * 4 / 8;   // 20000 tiles / 8 waves
    const int scat64Blocks = (int)(((long long)N_EDGES * 4 + B - 1) / B);

    // --- BN + layer 1 (feat 4 -> 64) -------------------------------------
    zero_kernel<<<(N_NODES * N_FEAT + B - 1) / B, B, 0, stream>>>(R1, N_NODES * N_FEAT);
    bn_kernel<<<(N_NODES * N_FEAT + B - 1) / B, B, 0, stream>>>(x, bn_g, bn_b, bn_m, bn_v, R0);
    scatter4_kernel<<<(N_EDGES + B - 1) / B, B, 0, stream>>>(R0, src, dst, R1);
    conv_gemm4_kernel<<<gemmBlocks, B, 0, stream>>>(R1, R0, W1_rel, b1_rel, W1_root, R2);

    // --- layer 2 (64 -> 64, ReLU) ----------------------------------------
    zero_kernel<<<(int)((NODEBUF + B - 1) / B), B, 0, stream>>>(R1, (int)NODEBUF);
    scatter64_kernel<<<scat64Blocks, B, 0, stream>>>(R2, src, dst, R1);
    conv_gemm64_kernel<<<gemmBlocks, B, 0, stream>>>(R1, R2, W2_rel, b2_rel, W2_root, R0, 1);

    // --- layer 3 (64 -> 64, no ReLU) -------------------------------------
    zero_kernel<<<(int)((NODEBUF + B - 1) / B), B, 0, stream>>>(R1, (int)NODEBUF);
    scatter64_kernel<<<scat64Blocks, B, 0, stream>>>(R0, src, dst, R1);
    conv_gemm64_kernel<<<gemmBlocks, B, 0, stream>>>(R1, R0, W3_rel, b3_rel, W3_root, R2, 0);

    // --- mean pool + final linear ----------------------------------------
    zero_kernel<<<1, B, 0, stream>>>(sums, N_GRAPHS * HIDDEN + N_GRAPHS);
    pool_kernel<<<(N_NODES * HIDDEN + B - 1) / B, B, 0, stream>>>(R2, batch, sums, cnts);
    final_kernel<<<1, 128, 0, stream>>>(sums, cnts, lin_W, lin_b, out);
}